// Kalman_Filter_24850680775215
// MI455X (gfx1250) — compile-verified
//
#include <hip/hip_runtime.h>
#include <cstddef>

#define T_LEN 512
#define BS_N  64
#define DA_N  16
#define DZ_N  32

// Only check for the WMMA builtin on the *device* pass; the host pass of the
// dual compilation never has AMDGCN builtins.
#if defined(__HIP_DEVICE_COMPILE__)
#if !__has_builtin(__builtin_amdgcn_wmma_f32_16x16x4_f32)
#error "device pass: missing __builtin_amdgcn_wmma_f32_16x16x4_f32"
#endif
#endif

typedef __attribute__((ext_vector_type(2))) float v2f;
typedef __attribute__((ext_vector_type(8))) float v8f;

// ---------------------------------------------------------------------------
// Generic small-matmul on LDS/global row-major buffers using f32 WMMA 16x16x4.
// Z[P,Q] = X[P,K] * Y[K,Q]   (TX/TY read X/Y transposed in-place).
// A-frag layout: lane = m + 16*h holds A[m, 4*kb+2h], A[m, 4*kb+2h+1]
// B-frag layout: lane = n + 16*h holds B[4*kb+2h, n], B[4*kb+2h+1, n]
// D layout:      VGPR v, lane = m + 16*h  ->  D[v + 8*h, m]
// Tile loop is wave-uniform -> EXEC stays all-ones around WMMA.
// ---------------------------------------------------------------------------
template<int P, int K, int Q, bool TX, bool TY>
__device__ __forceinline__
void mm_tiles(float* Z, int ldz, const float* X, int ldx,
              const float* Y, int ldy, int wave, int nwaves, int lane)
{
  const int m = lane & 15;
  const int h = lane >> 4;
  constexpr int TQ = Q / 16;
  constexpr int NT = (P / 16) * TQ;
  for (int tile = wave; tile < NT; tile += nwaves) {
    const int ti = (tile / TQ) * 16;
    const int tj = (tile % TQ) * 16;
    v8f acc = {0.f, 0.f, 0.f, 0.f, 0.f, 0.f, 0.f, 0.f};
#pragma unroll
    for (int kb = 0; kb < K / 4; ++kb) {
      const int k0 = 4 * kb + 2 * h;
      v2f av, bv;
      av.x = TX ? X[(k0    ) * ldx + (ti + m)] : X[(ti + m) * ldx + (k0    )];
      av.y = TX ? X[(k0 + 1) * ldx + (ti + m)] : X[(ti + m) * ldx + (k0 + 1)];
      bv.x = TY ? Y[(tj + m) * ldy + (k0    )] : Y[(k0    ) * ldy + (tj + m)];
      bv.y = TY ? Y[(tj + m) * ldy + (k0 + 1)] : Y[(k0 + 1) * ldy + (tj + m)];
#if defined(__HIP_DEVICE_COMPILE__)
      acc = __builtin_amdgcn_wmma_f32_16x16x4_f32(false, av, false, bv,
                                                  (short)0, acc, false, false);
#endif
    }
#pragma unroll
    for (int v = 0; v < 8; ++v)
      Z[(ti + v + 8 * h) * ldz + (tj + m)] = acc[v];
  }
}

// ---------------------------------------------------------------------------
// 16x16 Gauss-Jordan inverse, row-per-lane in registers, wave-internal shfl.
// S is SPD (C sig C^T + 0.1 I) -> no pivoting needed. Lanes 16..31 mirror
// rows 0..15 harmlessly; only lanes 0..15 write the result.
// ---------------------------------------------------------------------------
__device__ __forceinline__ void invert16_wave(const float* Sl, float* Si, int lane)
{
  const int r = lane & 15;
  float s[16], e[16];
#pragma unroll
  for (int j = 0; j < 16; ++j) { s[j] = Sl[r * 16 + j]; e[j] = (j == r) ? 1.f : 0.f; }
#pragma unroll
  for (int k = 0; k < 16; ++k) {
    float rk[16], ek[16];
#pragma unroll
    for (int j = 0; j < 16; ++j) { rk[j] = __shfl(s[j], k); ek[j] = __shfl(e[j], k); }
    const float pinv = 1.0f / rk[k];
    const float fk = s[k] * pinv;
    if (r == k) {
#pragma unroll
      for (int j = 0; j < 16; ++j) { s[j] = rk[j] * pinv; e[j] = ek[j] * pinv; }
    } else {
#pragma unroll
      for (int j = 0; j < 16; ++j) { s[j] -= fk * rk[j]; e[j] -= fk * ek[j]; }
    }
  }
  if (lane < 16) {
#pragma unroll
    for (int j = 0; j < 16; ++j) Si[lane * 16 + j] = e[j];
  }
}

__device__ __forceinline__ float sigmoidf_(float x) { return 1.f / (1.f + expf(-x)); }

// ---------------------------------------------------------------------------
// Kernel A: 2-layer LSTM (hidden=3) + softmax -> alpha[b,t,3].
// One wave per batch element; lanes 0..11 compute gates, 0..2 update state.
// ---------------------------------------------------------------------------
__global__ __launch_bounds__(32)
void lstm_alpha_kernel(const float* __restrict__ a,
                       const float* __restrict__ Wih0, const float* __restrict__ Whh0,
                       const float* __restrict__ bih0, const float* __restrict__ bhh0,
                       const float* __restrict__ Wih1, const float* __restrict__ Whh1,
                       const float* __restrict__ bih1, const float* __restrict__ bhh1,
                       float* __restrict__ alpha)
{
  const int b = blockIdx.x;
  const int L = threadIdx.x;
  __shared__ float w0[192], u0[36], w1[36], u1[36], bb0[12], bb1[12];
  __shared__ float x[16], h0[3], c0[3], h1[3], c1[3], g0[12], g1[12];

  for (int i = L; i < 192; i += 32) w0[i] = Wih0[i];
  for (int i = L; i < 36; i += 32) { u0[i] = Whh0[i]; w1[i] = Wih1[i]; u1[i] = Whh1[i]; }
  if (L < 12) { bb0[L] = bih0[L] + bhh0[L]; bb1[L] = bih1[L] + bhh1[L]; }
  if (L < 3) { h0[L] = 0.f; c0[L] = 0.f; h1[L] = 0.f; c1[L] = 0.f; }
  __syncthreads();

  for (int t = 0; t < T_LEN; ++t) {
    if (L < 16) x[L] = a[((size_t)b * T_LEN + t) * DA_N + L];
    __syncthreads();
    if (L < 12) {                                   // layer-0 gates (order i,f,g,o)
      float g = bb0[L];
      for (int j = 0; j < 16; ++j) g += w0[L * 16 + j] * x[j];
      for (int j = 0; j < 3; ++j)  g += u0[L * 3 + j] * h0[j];
      g0[L] = g;
    }
    __syncthreads();
    if (L < 3) {
      float ig = sigmoidf_(g0[L]),     fg = sigmoidf_(g0[3 + L]);
      float gg = tanhf(g0[6 + L]),     og = sigmoidf_(g0[9 + L]);
      float c = fg * c0[L] + ig * gg;
      c0[L] = c; h0[L] = og * tanhf(c);
    }
    __syncthreads();
    if (L < 12) {                                   // layer-1 gates
      float g = bb1[L];
      for (int j = 0; j < 3; ++j) g += w1[L * 3 + j] * h0[j] + u1[L * 3 + j] * h1[j];
      g1[L] = g;
    }
    __syncthreads();
    if (L < 3) {
      float ig = sigmoidf_(g1[L]),     fg = sigmoidf_(g1[3 + L]);
      float gg = tanhf(g1[6 + L]),     og = sigmoidf_(g1[9 + L]);
      float c = fg * c1[L] + ig * gg;
      c1[L] = c; h1[L] = og * tanhf(c);
    }
    __syncthreads();
    if (L == 0) {                                   // softmax over K=3
      float m = fmaxf(h1[0], fmaxf(h1[1], h1[2]));
      float e0 = expf(h1[0] - m), e1 = expf(h1[1] - m), e2 = expf(h1[2] - m);
      float inv = 1.f / (e0 + e1 + e2);
      float* ap = alpha + ((size_t)b * T_LEN + t) * 3;
      ap[0] = e0 * inv; ap[1] = e1 * inv; ap[2] = e2 * inv;
    }
    __syncthreads();
  }
}

// ---------------------------------------------------------------------------
// Kernel C: Kalman forward scan. One 128-thread block (4 waves) per batch
// element; all state LDS-resident; matmuls via f32 WMMA.
// Writes mu_f/sig_f straight into d_out (t=T-1 slots are final), and
// mu_p/sig_p into workspace for the smoother.
// ---------------------------------------------------------------------------
__global__ __launch_bounds__(128)
void kalman_forward_kernel(const float* __restrict__ a,
                           const float* __restrict__ Ag,
                           const float* __restrict__ Cg,
                           const float* __restrict__ Rg,
                           const float* __restrict__ Qg,
                           const float* __restrict__ mu0,
                           const float* __restrict__ sigma0,
                           const float* __restrict__ alpha,
                           float* __restrict__ mu_p_ws,
                           float* __restrict__ sig_p_ws,
                           float* __restrict__ out)
{
  const int b = blockIdx.x;
  const int tid = threadIdx.x;
  const int lane = tid & 31;
  const int wave = tid >> 5;

  __shared__ float Am[1024], Cm[512], sigp[1024], U[512];
  __shared__ float Sb[256], Si[256], Kg[512], W[1024], V[1024];
  __shared__ float muf[32], mup[32], munew[32], rv[16], al[4];

  for (int i = tid; i < 1024; i += 128) sigp[i] = sigma0[i];
  if (tid < 32) { muf[tid] = mu0[tid]; mup[tid] = mu0[tid]; }
  __syncthreads();

  for (int t = 0; t < T_LEN; ++t) {
    if (tid < 3) al[tid] = alpha[((size_t)b * T_LEN + t) * 3 + tid];
    __syncthreads();
    const float a0 = al[0], a1 = al[1], a2 = al[2];
    const float* A0 = Ag + ((size_t)0 * T_LEN + t) * 1024;
    const float* A1 = Ag + ((size_t)1 * T_LEN + t) * 1024;
    const float* A2 = Ag + ((size_t)2 * T_LEN + t) * 1024;
    for (int i = tid; i < 1024; i += 128) Am[i] = a0 * A0[i] + a1 * A1[i] + a2 * A2[i];
    const float* C0 = Cg + ((size_t)0 * T_LEN + t) * 512;
    const float* C1 = Cg + ((size_t)1 * T_LEN + t) * 512;
    const float* C2 = Cg + ((size_t)2 * T_LEN + t) * 512;
    for (int i = tid; i < 512; i += 128) Cm[i] = a0 * C0[i] + a1 * C1[i] + a2 * C2[i];
    if (t + 1 < T_LEN) {                            // prefetch next-step slices
      for (int i = tid; i < 192; i += 128) {
        int k = i >> 6, ln = i & 63;
        __builtin_prefetch(Ag + ((size_t)k * T_LEN + t + 1) * 1024 + (size_t)ln * 16, 0, 1);
      }
      for (int i = tid; i < 96; i += 128) {
        int k = i / 32, ln = i % 32;
        __builtin_prefetch(Cg + ((size_t)k * T_LEN + t + 1) * 512 + (size_t)ln * 16, 0, 1);
      }
    }
    __syncthreads();

    if (tid < 16) {                                 // r = a_t - C mu_f  (prev filtered)
      float s = a[((size_t)b * T_LEN + t) * DA_N + tid];
      for (int j = 0; j < 32; ++j) s -= Cm[tid * 32 + j] * muf[j];
      rv[tid] = s;
    }
    mm_tiles<32, 32, 16, false, true>(U, 16, sigp, 32, Cm, 32, wave, 4, lane);   // U = sig_p C^T
    __syncthreads();
    mm_tiles<16, 32, 16, false, false>(Sb, 16, Cm, 32, U, 16, wave, 4, lane);    // S = C U
    __syncthreads();
    for (int i = tid; i < 256; i += 128) Sb[i] += Rg[i];                          // S += R
    __syncthreads();
    if (wave == 0) invert16_wave(Sb, Si, lane);                                   // Si = S^-1
    __syncthreads();
    mm_tiles<32, 16, 16, false, false>(Kg, 16, U, 16, Si, 16, wave, 4, lane);    // Kg = U Si
    __syncthreads();
    mm_tiles<32, 16, 32, false, true>(W, 32, Kg, 16, U, 16, wave, 4, lane);      // W = Kg U^T
    __syncthreads();
    if (tid < 32) {                                                               // mu_new = mu_p + Kg r
      float s = mup[tid];
      for (int j = 0; j < 16; ++j) s += Kg[tid * 16 + j] * rv[j];
      munew[tid] = s;
    }
    for (int i = tid; i < 1024; i += 128) W[i] = sigp[i] - W[i];                  // sig_new
    __syncthreads();
    if (tid < 32) {                                                               // mu_p_new = A mu_new
      float s = 0.f;
      for (int j = 0; j < 32; ++j) s += Am[tid * 32 + j] * munew[j];
      mup[tid] = s;
    }
    mm_tiles<32, 32, 32, false, false>(V, 32, Am, 32, W, 32, wave, 4, lane);     // V = A sig_new
    __syncthreads();
    mm_tiles<32, 32, 32, false, true>(sigp, 32, V, 32, Am, 32, wave, 4, lane);   // sig_p = V A^T
    __syncthreads();

    float* ob = out + ((size_t)b * T_LEN + t) * 1056;
    for (int i = tid; i < 1024; i += 128) {
      float v = sigp[i] + Qg[i];                                                  // += Q (fused w/ store)
      sigp[i] = v;
      sig_p_ws[((size_t)t * BS_N + b) * 1024 + i] = v;
      ob[32 + i] = W[i];                                                          // sig_f -> out
    }
    if (tid < 32) {
      ob[tid] = munew[tid];                                                       // mu_f -> out
      muf[tid] = munew[tid];
      mu_p_ws[((size_t)t * BS_N + b) * 32 + tid] = mup[tid];
    }
    __syncthreads();
  }
}

// ---------------------------------------------------------------------------
// Kernel D: RTS smoother gains, fully parallel over (t,b), one wave each.
// J[t] = sig_f[t] * A[t+1]^T * inv(sig_p[t]); 32x32 GJ inverse in LDS.
// ---------------------------------------------------------------------------
__global__ __launch_bounds__(32)
void rts_gain_kernel(const float* __restrict__ Ag,
                     const float* __restrict__ alpha,
                     const float* __restrict__ sig_p_ws,
                     const float* __restrict__ out,     // filtered sig_f lives here
                     float* __restrict__ J_ws)
{
  const int idx = blockIdx.x;                 // (T-1)*BS blocks
  const int t = idx / BS_N;                   // 0..T-2
  const int b = idx - t * BS_N;
  const int L = threadIdx.x;

  __shared__ float P[1024], F[1024], Minv[1024], T2[1024];
  __shared__ float aug[32 * 65];

  const float a0 = alpha[((size_t)b * T_LEN + (t + 1)) * 3 + 0];
  const float a1 = alpha[((size_t)b * T_LEN + (t + 1)) * 3 + 1];
  const float a2 = alpha[((size_t)b * T_LEN + (t + 1)) * 3 + 2];
  const float* A0 = Ag + ((size_t)0 * T_LEN + t + 1) * 1024;
  const float* A1 = Ag + ((size_t)1 * T_LEN + t + 1) * 1024;
  const float* A2 = Ag + ((size_t)2 * T_LEN + t + 1) * 1024;
  for (int i = L; i < 1024; i += 32) P[i] = a0 * A0[i] + a1 * A1[i] + a2 * A2[i];
  const float* fb = out + ((size_t)b * T_LEN + t) * 1056 + 32;
  for (int i = L; i < 1024; i += 32) F[i] = fb[i];
  const float* sp = sig_p_ws + ((size_t)t * BS_N + b) * 1024;
  for (int i2 = 0; i2 < 32; ++i2) {
    aug[i2 * 65 + L]      = sp[i2 * 32 + L];
    aug[i2 * 65 + 32 + L] = (i2 == L) ? 1.f : 0.f;
  }
  __syncthreads();
  for (int k = 0; k < 32; ++k) {              // Gauss-Jordan (SPD, no pivoting)
    const float pinv = 1.0f / aug[k * 65 + k];
    __syncthreads();
    aug[k * 65 + L]      *= pinv;             // scale pivot row (lane = column)
    aug[k * 65 + 32 + L] *= pinv;
    __syncthreads();
    const float f = (L == k) ? 0.f : aug[L * 65 + k];   // lane = row
    for (int j = 0; j < 64; ++j)
      aug[L * 65 + j] -= f * aug[k * 65 + j];
    __syncthreads();
  }
  for (int i2 = 0; i2 < 32; ++i2) Minv[i2 * 32 + L] = aug[i2 * 65 + 32 + L];
  __syncthreads();
  mm_tiles<32, 32, 32, true,  false>(T2, 32, P, 32, Minv, 32, 0, 1, L);   // T2 = A^T inv
  __syncthreads();
  mm_tiles<32, 32, 32, false, false>(J_ws + ((size_t)t * BS_N + b) * 1024, 32,
                                     F, 32, T2, 32, 0, 1, L);             // J = sig_f T2 -> global
}

// ---------------------------------------------------------------------------
// Kernel E: backward smoothing scan. One 64-thread block per batch element.
// sig_s = sig_f + J (sig_next - sig_p); mu_s = mu_f + J (mu_next - mu_p).
// Reads filtered values from d_out and overwrites them in place.
// ---------------------------------------------------------------------------
__global__ __launch_bounds__(64)
void rts_scan_kernel(const float* __restrict__ mu_p_ws,
                     const float* __restrict__ sig_p_ws,
                     const float* __restrict__ J_ws,
                     float* __restrict__ out)
{
  const int b = blockIdx.x;
  const int tid = threadIdx.x;
  const int lane = tid & 31;
  const int wave = tid >> 5;
  __shared__ float Jl[1024], Dm[1024], Sn[1024];
  __shared__ float mun[32], mudn[32];

  float* outb = out + (size_t)b * T_LEN * 1056;
  for (int i = tid; i < 1024; i += 64) Sn[i] = outb[(size_t)(T_LEN - 1) * 1056 + 32 + i];
  if (tid < 32) mun[tid] = outb[(size_t)(T_LEN - 1) * 1056 + tid];
  __syncthreads();

  for (int t = T_LEN - 2; t >= 0; --t) {
    const float* Jg = J_ws + ((size_t)t * BS_N + b) * 1024;
    const float* sp = sig_p_ws + ((size_t)t * BS_N + b) * 1024;
    for (int i = tid; i < 1024; i += 64) { Jl[i] = Jg[i]; Dm[i] = Sn[i] - sp[i]; }
    if (tid < 32) mudn[tid] = mun[tid] - mu_p_ws[((size_t)t * BS_N + b) * 32 + tid];
    __syncthreads();
    mm_tiles<32, 32, 32, false, false>(Sn, 32, Jl, 32, Dm, 32, wave, 2, lane);  // J * diff
    __syncthreads();
    float* ob = outb + (size_t)t * 1056;
    for (int i = tid; i < 1024; i += 64) {
      float v = ob[32 + i] + Sn[i];       // sig_s = sig_f + J diff
      Sn[i] = v;
      ob[32 + i] = v;
    }
    if (tid < 32) {
      float s = 0.f;
      for (int j = 0; j < 32; ++j) s += Jl[tid * 32 + j] * mudn[j];
      float v = ob[tid] + s;              // mu_s = mu_f + J dmu
      mun[tid] = v;
      ob[tid] = v;
    }
    __syncthreads();
  }
}

// ---------------------------------------------------------------------------
// Workspace layout (floats):
//   alpha : BS*T*3            =     98,304
//   mu_p  : T*BS*32           =  1,048,576
//   sig_p : T*BS*1024         = 33,554,432
//   J     : (T-1)*BS*1024     = 33,488,896
// total ~273 MB.
// ---------------------------------------------------------------------------
extern "C" void kernel_launch(void* const* d_in, const int* in_sizes, int n_in,
                              void* d_out, int out_size, void* d_ws, size_t ws_size,
                              hipStream_t stream)
{
  const float* a      = (const float*)d_in[0];
  const float* Ag     = (const float*)d_in[1];
  const float* Cg     = (const float*)d_in[2];
  const float* Rg     = (const float*)d_in[3];
  const float* Qg     = (const float*)d_in[4];
  const float* mu0    = (const float*)d_in[5];
  const float* sigma0 = (const float*)d_in[6];
  const float* Wih0   = (const float*)d_in[7];
  const float* Whh0   = (const float*)d_in[8];
  const float* bih0   = (const float*)d_in[9];
  const float* bhh0   = (const float*)d_in[10];
  const float* Wih1   = (const float*)d_in[11];
  const float* Whh1   = (const float*)d_in[12];
  const float* bih1   = (const float*)d_in[13];
  const float* bhh1   = (const float*)d_in[14];
  float* out = (float*)d_out;
  float* ws  = (float*)d_ws;
  (void)in_sizes; (void)n_in; (void)out_size; (void)ws_size;

  const size_t n_alpha = (size_t)BS_N * T_LEN * 3;
  const size_t n_mup   = (size_t)T_LEN * BS_N * 32;
  const size_t n_sigp  = (size_t)T_LEN * BS_N * 1024;
  float* alpha    = ws;
  float* mu_p_ws  = alpha + n_alpha;
  float* sig_p_ws = mu_p_ws + n_mup;
  float* J_ws     = sig_p_ws + n_sigp;

  lstm_alpha_kernel<<<BS_N, 32, 0, stream>>>(a, Wih0, Whh0, bih0, bhh0,
                                             Wih1, Whh1, bih1, bhh1, alpha);
  kalman_forward_kernel<<<BS_N, 128, 0, stream>>>(a, Ag, Cg, Rg, Qg, mu0, sigma0,
                                                  alpha, mu_p_ws, sig_p_ws, out);
  rts_gain_kernel<<<(T_LEN - 1) * BS_N, 32, 0, stream>>>(Ag, alpha, sig_p_ws, out, J_ws);
  rts_scan_kernel<<<BS_N, 64, 0, stream>>>(mu_p_ws, sig_p_ws, J_ws, out);
}